// MambaHead_37641093382895
// MI455X (gfx1250) — compile-verified
//
#include <hip/hip_runtime.h>
#include <math.h>

#define D_MODEL   512
#define D_STATE   16
#define D_INNER   1024
#define D_CONV    4
#define DT_RANK   32
#define N_LAYERS  4
#define BATCH     2
#define SEQ       1024
#define TOK       (BATCH * SEQ)          // 2048 tokens
#define LABEL_LEN 3
#define LABEL_DIM 2
#define EPS       1e-5f
#define CHUNKS    16                     // chunked-scan chunks per sequence
#define LC        (SEQ / CHUNKS)         // 64 steps per chunk

typedef float v2f __attribute__((ext_vector_type(2)));
typedef float v4f __attribute__((ext_vector_type(4)));
typedef float v8f __attribute__((ext_vector_type(8)));

// ---------------------------------------------------------------------------
// CDNA5 fp32 WMMA: D(16x16,f32) = A(16x4,f32) * B(4x16,f32) + C
// Natural fragment: lane -> row/col = lane&15, K pair at 2*(lane>>4).
// We exploit K-permutation freedom: any K->slot assignment is valid as long
// as A and B use the identical assignment (partial sums accumulate via C).
// Each lane loads 4 consecutive floats at k + 4*(lane>>4) with ONE b128 load
// and feeds two WMMAs (.xy covers K{k,k+1,k+4,k+5}, .zw covers {k+2,k+3,
// k+6,k+7}) -- every K index covered exactly once, identically on A and B.
// ---------------------------------------------------------------------------
__device__ __forceinline__ v8f wmma4(v2f a, v2f b, v8f c) {
  return __builtin_amdgcn_wmma_f32_16x16x4_f32(false, a, false, b,
                                               (short)0, c, false, false);
}
__device__ __forceinline__ void wmma8(v4f a, v4f b, v8f& acc) {
  v2f alo = __builtin_shufflevector(a, a, 0, 1);
  v2f ahi = __builtin_shufflevector(a, a, 2, 3);
  v2f blo = __builtin_shufflevector(b, b, 0, 1);
  v2f bhi = __builtin_shufflevector(b, b, 2, 3);
  acc = wmma4(alo, blo, acc);
  acc = wmma4(ahi, bhi, acc);
}

__device__ __forceinline__ float softplus_f(float x) {
  // matches jax.nn.softplus = logaddexp(x, 0)
  return fmaxf(x, 0.f) + log1pf(expf(-fabsf(x)));
}
__device__ __forceinline__ float silu_f(float x) {
  return x / (1.f + expf(-x));
}

// ---------------------------------------------------------------------------
// GEMM: C[M,N] (row-major, ldc) = A[M,K] (row stride lda) * W[N,K]^T
// Block = (32,4): 4 waves, each wave owns a 32(M) x 64(N) tile
// (2 A-frags x 4 B-frags = 8 accumulators).  Inner loop (unroll 1, software
// pipelined): 6 b128 prefetch loads overlap 16 WMMAs on previous registers.
// EPI 0: store.  EPI 1: softplus(x + bias[n]).  EPI 2: C += x (residual).
// Requires M%128==0, N%64==0, K%8==0, all threads active (EXEC all-1).
// ---------------------------------------------------------------------------
template <int EPI>
__global__ __launch_bounds__(128)
void gemm_wmma_kernel(const float* __restrict__ A, int lda,
                      const float* __restrict__ W,
                      float* __restrict__ C, int ldc,
                      const float* __restrict__ bias, int K) {
  const int lane = threadIdx.x;
  const int r    = lane & 15;
  const int g    = lane >> 4;
  const int m0   = (blockIdx.y * 4 + threadIdx.y) * 32;
  const int n0   = blockIdx.x * 64;

  const float* Ap0 = A + (size_t)(m0 + r) * lda + 4 * g;
  const float* Ap1 = Ap0 + (size_t)16 * lda;
  const float* Wp0 = W + (size_t)(n0 + r) * K + 4 * g;
  const float* Wp1 = Wp0 + (size_t)16 * K;
  const float* Wp2 = Wp0 + (size_t)32 * K;
  const float* Wp3 = Wp0 + (size_t)48 * K;

  v8f acc[2][4] = {};

  // prologue: fragment group for k = 0
  v4f a0 = *(const v4f*)(Ap0);
  v4f a1 = *(const v4f*)(Ap1);
  v4f b0 = *(const v4f*)(Wp0);
  v4f b1 = *(const v4f*)(Wp1);
  v4f b2 = *(const v4f*)(Wp2);
  v4f b3 = *(const v4f*)(Wp3);

#pragma unroll 1
  for (int k = 8; k < K; k += 8) {
    // prefetch next 8-K step while current 16 WMMAs execute
    v4f na0 = *(const v4f*)(Ap0 + k);
    v4f na1 = *(const v4f*)(Ap1 + k);
    v4f nb0 = *(const v4f*)(Wp0 + k);
    v4f nb1 = *(const v4f*)(Wp1 + k);
    v4f nb2 = *(const v4f*)(Wp2 + k);
    v4f nb3 = *(const v4f*)(Wp3 + k);

    wmma8(a0, b0, acc[0][0]);
    wmma8(a1, b0, acc[1][0]);
    wmma8(a0, b1, acc[0][1]);
    wmma8(a1, b1, acc[1][1]);
    wmma8(a0, b2, acc[0][2]);
    wmma8(a1, b2, acc[1][2]);
    wmma8(a0, b3, acc[0][3]);
    wmma8(a1, b3, acc[1][3]);

    a0 = na0; a1 = na1; b0 = nb0; b1 = nb1; b2 = nb2; b3 = nb3;
  }
  // epilogue 8-K step
  wmma8(a0, b0, acc[0][0]);
  wmma8(a1, b0, acc[1][0]);
  wmma8(a0, b1, acc[0][1]);
  wmma8(a1, b1, acc[1][1]);
  wmma8(a0, b2, acc[0][2]);
  wmma8(a1, b2, acc[1][2]);
  wmma8(a0, b3, acc[0][3]);
  wmma8(a1, b3, acc[1][3]);

#pragma unroll
  for (int s = 0; s < 2; ++s) {
    const int rowb = m0 + s * 16 + 8 * g;
#pragma unroll
    for (int t = 0; t < 4; ++t) {
      const int col = n0 + t * 16 + r;
#pragma unroll
      for (int v = 0; v < 8; ++v) {
        const size_t off = (size_t)(rowb + v) * ldc + col;
        float val = acc[s][t][v];
        if (EPI == 1)      C[off]  = softplus_f(val + bias[col]);
        else if (EPI == 2) C[off] += val;
        else               C[off]  = val;
      }
    }
  }
}

// ---------------------------------------------------------------------------
// RMSNorm: one wave per token (512 elems -> 16 per lane).
// ---------------------------------------------------------------------------
__global__ __launch_bounds__(32)
void rmsnorm_kernel(const float* __restrict__ x, const float* __restrict__ w,
                    float* __restrict__ o) {
  const int t    = blockIdx.x;
  const int lane = threadIdx.x;
  const float* xr = x + (size_t)t * D_MODEL;
  float v[16];
  float ss = 0.f;
#pragma unroll
  for (int i = 0; i < 16; ++i) {
    v[i] = xr[lane + 32 * i];
    ss += v[i] * v[i];
  }
#pragma unroll
  for (int off = 16; off > 0; off >>= 1) ss += __shfl_xor(ss, off, 32);
  const float s = rsqrtf(ss * (1.f / D_MODEL) + EPS);
  float* orow = o + (size_t)t * D_MODEL;
#pragma unroll
  for (int i = 0; i < 16; ++i)
    orow[lane + 32 * i] = v[i] * s * w[lane + 32 * i];
}

// ---------------------------------------------------------------------------
// Causal depthwise conv (width 4) over xz[:, :D_INNER] + bias + SiLU -> xi
// ---------------------------------------------------------------------------
__global__ __launch_bounds__(256)
void conv_silu_kernel(const float* __restrict__ xz, const float* __restrict__ cw,
                      const float* __restrict__ cb, float* __restrict__ xi) {
  const int idx = blockIdx.x * 256 + threadIdx.x;  // t * D_INNER + e
  const int e = idx & (D_INNER - 1);
  const int t = idx >> 10;
  const int l = t & (SEQ - 1);
  float s = cb[e];
  const float* wv = cw + e * D_CONV;
#pragma unroll
  for (int j = 0; j < D_CONV; ++j) {
    const int ls = l - (D_CONV - 1) + j;
    if (ls >= 0)
      s += xz[(size_t)(t - (D_CONV - 1) + j) * (2 * D_INNER) + e] * wv[j];
  }
  xi[idx] = silu_f(s);
}

// ---------------------------------------------------------------------------
// Chunked selective scan.  Recurrence h_l = a_l*h_{l-1} + bx_l is linear and
// elementwise per state n, so split L into CHUNKS chunks:
//   Pass A: per (b,chunk,e) compute P_n = prod(a) and q_n = scan-from-zero.
//   Pass B: per (b,e,n) serial combine over the 16 chunks -> chunk entry h.
//   Pass C: per (b,chunk,e) re-run chunk from its entry state, emit fused
//           y = (h.C + Dp*xi) * silu(z).
// Thread mapping keeps e fastest -> delta/xi/z coalesced, Bm/Cm broadcast.
// ---------------------------------------------------------------------------
__global__ __launch_bounds__(256)
void scan_passA(const float* __restrict__ delta, const float* __restrict__ xi,
                const float* __restrict__ dbc, const float* __restrict__ A_log,
                float* __restrict__ Pbuf, float* __restrict__ Qbuf) {
  const int tid = blockIdx.x * 256 + threadIdx.x;  // (b*CHUNKS + c)*E + e
  const int e  = tid & (D_INNER - 1);
  const int bc = tid >> 10;
  const int c  = bc & (CHUNKS - 1);
  const int b  = bc >> 4;

  float Ae[D_STATE], h[D_STATE], P[D_STATE];
#pragma unroll
  for (int n = 0; n < D_STATE; ++n) {
    Ae[n] = -expf(A_log[e * D_STATE + n]);
    h[n]  = 0.f;
    P[n]  = 1.f;
  }
  const int l0 = c * LC;
  for (int l = l0; l < l0 + LC; ++l) {
    const int t = b * SEQ + l;
    const float d  = delta[(size_t)t * D_INNER + e];
    const float xv = xi[(size_t)t * D_INNER + e];
    const float dx = d * xv;
    const float* bcp = dbc + (size_t)t * (DT_RANK + 2 * D_STATE);
#pragma unroll
    for (int n = 0; n < D_STATE; ++n) {
      const float a = expf(d * Ae[n]);
      h[n] = a * h[n] + dx * bcp[DT_RANK + n];
      P[n] *= a;
    }
  }
  const size_t o = (((size_t)(b * D_INNER + e)) * CHUNKS + c) * D_STATE;
#pragma unroll
  for (int n = 0; n < D_STATE; ++n) {
    Pbuf[o + n] = P[n];
    Qbuf[o + n] = h[n];
  }
}

__global__ __launch_bounds__(256)
void scan_passB(const float* __restrict__ Pbuf, const float* __restrict__ Qbuf,
                float* __restrict__ Hstart) {
  const int tid = blockIdx.x * 256 + threadIdx.x;  // (b*E + e)*16 + n
  const size_t base = (size_t)(tid >> 4) * (CHUNKS * D_STATE) + (tid & 15);
  float hs = 0.f;
#pragma unroll
  for (int c = 0; c < CHUNKS; ++c) {
    const size_t o = base + (size_t)c * D_STATE;
    Hstart[o] = hs;
    hs = Pbuf[o] * hs + Qbuf[o];
  }
}

__global__ __launch_bounds__(256)
void scan_passC(const float* __restrict__ delta, const float* __restrict__ xi,
                const float* __restrict__ dbc, const float* __restrict__ xz,
                const float* __restrict__ A_log, const float* __restrict__ Dp,
                const float* __restrict__ Hstart, float* __restrict__ y) {
  const int tid = blockIdx.x * 256 + threadIdx.x;  // (b*CHUNKS + c)*E + e
  const int e  = tid & (D_INNER - 1);
  const int bc = tid >> 10;
  const int c  = bc & (CHUNKS - 1);
  const int b  = bc >> 4;

  float Ae[D_STATE], h[D_STATE];
  const size_t hsb = (((size_t)(b * D_INNER + e)) * CHUNKS + c) * D_STATE;
#pragma unroll
  for (int n = 0; n < D_STATE; ++n) {
    Ae[n] = -expf(A_log[e * D_STATE + n]);
    h[n]  = Hstart[hsb + n];
  }
  const float dpe = Dp[e];
  const int l0 = c * LC;
  for (int l = l0; l < l0 + LC; ++l) {
    const int t = b * SEQ + l;
    const float d  = delta[(size_t)t * D_INNER + e];
    const float xv = xi[(size_t)t * D_INNER + e];
    const float dx = d * xv;
    const float* bcp = dbc + (size_t)t * (DT_RANK + 2 * D_STATE);
    float acc = 0.f;
#pragma unroll
    for (int n = 0; n < D_STATE; ++n) {
      const float hn = expf(d * Ae[n]) * h[n] + dx * bcp[DT_RANK + n];
      h[n] = hn;
      acc += hn * bcp[DT_RANK + D_STATE + n];
    }
    const float zv = xz[(size_t)t * (2 * D_INNER) + D_INNER + e];
    y[(size_t)t * D_INNER + e] = (acc + dpe * xv) * silu_f(zv);
  }
}

// ---------------------------------------------------------------------------
// Head: out[b,l,k] = x_cur[b, SEQ-3+l, :] . head_w[k,:] + head_b[k]
// One wave per output scalar (12 total).
// ---------------------------------------------------------------------------
__global__ __launch_bounds__(32)
void head_kernel(const float* __restrict__ xc, const float* __restrict__ hw,
                 const float* __restrict__ hb, float* __restrict__ out) {
  const int o   = blockIdx.x;                       // b*6 + l*2 + k
  const int b   = o / (LABEL_LEN * LABEL_DIM);
  const int rem = o % (LABEL_LEN * LABEL_DIM);
  const int l   = rem / LABEL_DIM;
  const int k   = rem % LABEL_DIM;
  const int t   = b * SEQ + (SEQ - LABEL_LEN) + l;
  float s = 0.f;
  for (int i = threadIdx.x; i < D_MODEL; i += 32)
    s += xc[(size_t)t * D_MODEL + i] * hw[k * D_MODEL + i];
#pragma unroll
  for (int off = 16; off > 0; off >>= 1) s += __shfl_xor(s, off, 32);
  if (threadIdx.x == 0) out[o] = s + hb[k];
}

// ---------------------------------------------------------------------------
extern "C" void kernel_launch(void* const* d_in, const int* in_sizes, int n_in,
                              void* d_out, int out_size, void* d_ws, size_t ws_size,
                              hipStream_t stream) {
  const float* x       = (const float*)d_in[0];
  const float* norm_w  = (const float*)d_in[1];
  const float* in_w    = (const float*)d_in[2];
  const float* conv_w  = (const float*)d_in[3];
  const float* conv_b  = (const float*)d_in[4];
  const float* xproj_w = (const float*)d_in[5];
  const float* dt_w    = (const float*)d_in[6];
  const float* dt_b    = (const float*)d_in[7];
  const float* A_log   = (const float*)d_in[8];
  const float* Dp      = (const float*)d_in[9];
  const float* out_w   = (const float*)d_in[10];
  const float* head_w  = (const float*)d_in[11];
  const float* head_b  = (const float*)d_in[12];

  float* ws     = (float*)d_ws;
  float* x_cur  = ws;                                     // TOK*512
  float* xnorm  = ws + (size_t)TOK * D_MODEL;             // TOK*512
  float* xz     = xnorm + (size_t)TOK * D_MODEL;          // TOK*2048
  float* xi     = xz    + (size_t)TOK * 2 * D_INNER;      // TOK*1024
  float* dbc    = xi    + (size_t)TOK * D_INNER;          // TOK*64
  float* delta  = dbc   + (size_t)TOK * (DT_RANK + 2 * D_STATE); // TOK*1024
  float* ybuf   = delta + (size_t)TOK * D_INNER;          // TOK*1024
  float* Pbuf   = ybuf  + (size_t)TOK * D_INNER;          // B*E*C*16
  float* Qbuf   = Pbuf  + (size_t)BATCH * D_INNER * CHUNKS * D_STATE;
  float* Hstart = Qbuf  + (size_t)BATCH * D_INNER * CHUNKS * D_STATE;

  // residual stream init (capture-safe stream-ordered d2d copy)
  hipMemcpyAsync(x_cur, x, sizeof(float) * (size_t)TOK * D_MODEL,
                 hipMemcpyDeviceToDevice, stream);

  const dim3 gblk(32, 4);
  const int scan_threads = BATCH * CHUNKS * D_INNER;      // 32768
  for (int i = 0; i < N_LAYERS; ++i) {
    rmsnorm_kernel<<<TOK, 32, 0, stream>>>(x_cur, norm_w + i * D_MODEL, xnorm);

    // xz = xnorm @ in_w^T   (M=2048, N=2048, K=512)
    gemm_wmma_kernel<0><<<dim3(2 * D_INNER / 64, TOK / 128), gblk, 0, stream>>>(
        xnorm, D_MODEL, in_w + (size_t)i * 2 * D_INNER * D_MODEL,
        xz, 2 * D_INNER, nullptr, D_MODEL);

    conv_silu_kernel<<<TOK * D_INNER / 256, 256, 0, stream>>>(
        xz, conv_w + i * D_INNER * D_CONV, conv_b + i * D_INNER, xi);

    // dbc = xi @ xproj_w^T  (M=2048, N=64, K=1024)
    gemm_wmma_kernel<0><<<dim3(1, TOK / 128), gblk, 0, stream>>>(
        xi, D_INNER, xproj_w + (size_t)i * (DT_RANK + 2 * D_STATE) * D_INNER,
        dbc, DT_RANK + 2 * D_STATE, nullptr, D_INNER);

    // delta = softplus(dbc[:, :32] @ dt_w^T + dt_b)  (M=2048, N=1024, K=32)
    gemm_wmma_kernel<1><<<dim3(D_INNER / 64, TOK / 128), gblk, 0, stream>>>(
        dbc, DT_RANK + 2 * D_STATE, dt_w + (size_t)i * D_INNER * DT_RANK,
        delta, D_INNER, dt_b + i * D_INNER, DT_RANK);

    // chunked scan (3 passes)
    scan_passA<<<scan_threads / 256, 256, 0, stream>>>(
        delta, xi, dbc, A_log + (size_t)i * D_INNER * D_STATE, Pbuf, Qbuf);
    scan_passB<<<BATCH * D_INNER * D_STATE / 256, 256, 0, stream>>>(
        Pbuf, Qbuf, Hstart);
    scan_passC<<<scan_threads / 256, 256, 0, stream>>>(
        delta, xi, dbc, xz, A_log + (size_t)i * D_INNER * D_STATE,
        Dp + i * D_INNER, Hstart, ybuf);

    // x_cur += ybuf @ out_w^T  (M=2048, N=512, K=1024)
    gemm_wmma_kernel<2><<<dim3(D_MODEL / 64, TOK / 128), gblk, 0, stream>>>(
        ybuf, D_INNER, out_w + (size_t)i * D_MODEL * D_INNER,
        x_cur, D_MODEL, nullptr, D_INNER);
  }

  head_kernel<<<BATCH * LABEL_LEN * LABEL_DIM, 32, 0, stream>>>(
      x_cur, head_w, head_b, (float*)d_out);
}